// KAN_layer_6854767805283
// MI455X (gfx1250) — compile-verified
//
#include <hip/hip_runtime.h>

typedef float v2f __attribute__((ext_vector_type(2)));
typedef float v8f __attribute__((ext_vector_type(8)));

#define IN_DIM  64
#define OUT_DIM 64
#define HID     64
#define DDIM    (IN_DIM * OUT_DIM)   // 4096 independent edge-MLPs
#define BSZ     256                  // batch
#define LSTR    68                   // LDS row stride (floats): bank-conflict-free for b64 frag loads

// One workgroup per edge d=(o,i). 256 threads = 8 wave32.
// Activations (256x64) live entirely in LDS; weights streamed from HBM once.
__launch_bounds__(256)
__global__ void kan_edge_kernel(const float* __restrict__ x,
                                const float* __restrict__ w0, const float* __restrict__ b0,
                                const float* __restrict__ w1, const float* __restrict__ b1,
                                const float* __restrict__ w2, const float* __restrict__ b2,
                                const float* __restrict__ wout,
                                float* __restrict__ outbuf, int use_atomic)
{
    __shared__ float act0[BSZ * LSTR];   // 69632 B
    __shared__ float act1[BSZ * LSTR];   // 69632 B
    __shared__ float xs[BSZ];
    __shared__ float ws0[HID], bs0[HID], wos[HID];

    const int d   = blockIdx.x;
    const int o   = d >> 6;
    const int i   = d & 63;
    const int tid = threadIdx.x;

    // ---- stage x column + per-edge vectors (single barrier covers all) ----
    xs[tid & 255] = x[(tid & 255) * IN_DIM + i];
    if (tid < HID) {
        ws0[tid] = w0[d * HID + tid];
        bs0[tid] = b0[d * HID + tid];
        wos[tid] = wout[o * (IN_DIM * HID) + i * HID + tid];
    }
    __syncthreads();

    // ---- input layer: act0[b][h] = relu(x[b,i]*w0[d,h] + b0[d,h]) ----
    {
        const int   h  = tid & 63;
        const int   bg = tid >> 6;
        const float wv = ws0[h];
        const float bv = bs0[h];
        for (int b = bg; b < BSZ; b += 4) {
            float v = fmaf(xs[b], wv, bv);
            act0[b * LSTR + h] = v > 0.f ? v : 0.f;
        }
    }
    __syncthreads();

    const int lane  = tid & 31;
    const int r     = lane & 15;     // row-in-tile index
    const int half  = lane >> 4;     // K/M half selector per WMMA f32 layout
    const int nbase = ((tid >> 5) & 3) * 16;  // wave's N tile
    const int mhalf = (tid >> 5) >> 2;        // wave's half of M tiles

    // ---- one hidden layer: dst = relu(src @ W^T + bias), all 16x16x4 f32 WMMA ----
    auto layer = [&](const float* __restrict__ src, float* __restrict__ dst,
                     const float* __restrict__ W, const float* __restrict__ Bv) {
        // B panel (16 N-rows x 64 K) resident in 32 VGPRs, loaded once from HBM.
        // B[k, n] = W[n*64 + k]; lane n=r holds K pair {c*4+2h, c*4+2h+1} in Bfrag[c].
        v2f Bfrag[16];
        const float* wrow = W + (size_t)(nbase + r) * HID + 2 * half;
        #pragma unroll
        for (int c = 0; c < 16; ++c)
            Bfrag[c] = *(const v2f*)(wrow + c * 4);
        const float bias = Bv[nbase + r];   // bias indexed by output column

        #pragma unroll
        for (int t = 0; t < 4; ++t) {
            const int m0 = (mhalf * 8 + 2 * t) * 16;
            const int m1 = m0 + 16;
            v8f acc0 = {0.f,0.f,0.f,0.f,0.f,0.f,0.f,0.f};
            v8f acc1 = {0.f,0.f,0.f,0.f,0.f,0.f,0.f,0.f};
            const float* a0p = src + (m0 + r) * LSTR + 2 * half;
            const float* a1p = src + (m1 + r) * LSTR + 2 * half;
            #pragma unroll
            for (int c = 0; c < 16; ++c) {
                v2f a0 = *(const v2f*)(a0p + c * 4);
                v2f a1 = *(const v2f*)(a1p + c * 4);
                acc0 = __builtin_amdgcn_wmma_f32_16x16x4_f32(
                           false, a0, false, Bfrag[c], (short)0, acc0, false, false);
                acc1 = __builtin_amdgcn_wmma_f32_16x16x4_f32(
                           false, a1, false, Bfrag[c], (short)0, acc1, false, false);
            }
            #pragma unroll
            for (int v = 0; v < 8; ++v) {
                const int row0 = m0 + v + 8 * half;   // C/D layout: M = v (+8 for upper half)
                const int row1 = m1 + v + 8 * half;
                float v0 = acc0[v] + bias; v0 = v0 > 0.f ? v0 : 0.f;
                float v1 = acc1[v] + bias; v1 = v1 > 0.f ? v1 : 0.f;
                dst[row0 * LSTR + nbase + r] = v0;
                dst[row1 * LSTR + nbase + r] = v1;
            }
        }
    };

    layer(act0, act1, w1 + (size_t)d * HID * HID, b1 + d * HID);
    __syncthreads();
    layer(act1, act0, w2 + (size_t)d * HID * HID, b2 + d * HID);
    __syncthreads();

    // ---- output contraction for this edge: partial[b] = dot(wout[o, i*64:], h2[b,:]) ----
    {
        const int b = tid;
        float s = 0.f;
        const float* row = act0 + b * LSTR;
        #pragma unroll 8
        for (int h = 0; h < HID; ++h) s = fmaf(wos[h], row[h], s);
        if (use_atomic) {
            __hip_atomic_fetch_add(&outbuf[b * OUT_DIM + o], s,
                                   __ATOMIC_RELAXED, __HIP_MEMORY_SCOPE_AGENT);
        } else {
            outbuf[(size_t)d * BSZ + b] = s;  // partials workspace, reduced below
        }
    }
}

// Deterministic reduction over the 64 input edges of each output neuron.
__global__ void kan_reduce_kernel(const float* __restrict__ part,
                                  const float* __restrict__ bout,
                                  float* __restrict__ out)
{
    const int o = blockIdx.x;    // 64
    const int b = threadIdx.x;   // 256
    float s = bout[o];
    const float* p = part + (size_t)o * 64 * BSZ + b;
    #pragma unroll 8
    for (int i = 0; i < 64; ++i) s += p[i * BSZ];
    out[b * OUT_DIM + o] = s;
}

// Atomic fallback path: seed out with bias before edge kernel accumulates.
__global__ void kan_init_out(const float* __restrict__ bout, float* __restrict__ out)
{
    const int idx = blockIdx.x * 256 + threadIdx.x;  // 16384 = 256*64
    out[idx] = bout[idx & 63];
}

extern "C" void kernel_launch(void* const* d_in, const int* in_sizes, int n_in,
                              void* d_out, int out_size, void* d_ws, size_t ws_size,
                              hipStream_t stream)
{
    const float* x    = (const float*)d_in[0];
    const float* w0   = (const float*)d_in[1];
    const float* b0   = (const float*)d_in[2];
    const float* w1   = (const float*)d_in[3];
    const float* b1   = (const float*)d_in[4];
    const float* w2   = (const float*)d_in[5];
    const float* b2   = (const float*)d_in[6];
    const float* wout = (const float*)d_in[7];
    const float* bout = (const float*)d_in[8];
    float* out = (float*)d_out;

    const size_t need = (size_t)DDIM * BSZ * sizeof(float);  // 4 MB partials
    if (ws_size >= need) {
        float* part = (float*)d_ws;
        kan_edge_kernel<<<DDIM, 256, 0, stream>>>(x, w0, b0, w1, b1, w2, b2, wout, part, 0);
        kan_reduce_kernel<<<OUT_DIM, BSZ, 0, stream>>>(part, bout, out);
    } else {
        kan_init_out<<<64, 256, 0, stream>>>(bout, out);
        kan_edge_kernel<<<DDIM, 256, 0, stream>>>(x, w0, b0, w1, b1, w2, b2, wout, out, 1);
    }
}